// OFModelST_9552007266823
// MI455X (gfx1250) — compile-verified
//
#include <hip/hip_runtime.h>
#include <math.h>

typedef float v2f __attribute__((ext_vector_type(2)));
typedef float v8f __attribute__((ext_vector_type(8)));

#define LAMBD 0.01f

// ---------- sizes ----------
#define NC_H 51200      // 10 frames * 2 variants * 16 batch * 160 cols
#define NC_V 40960      // 10 * 2 * 16 * 128
#define NC_T 377856     // 16 * 23616
#define KPAD 48
#define FEAT 23616      // 41*(160+128)*2
#define OFF_H0 0
#define OFF_V0 6560     // 41*160
#define OFF_H1 11808    // +41*128
#define OFF_V1 18368    // +41*160

// ---------------------------------------------------------------------------
// Dictionary construction: make_dict(rows, rr, theta) column-normalized [rows,41]
// d=0: dicH(16, rrSH,thSH)  d=1: dicV(20, rrSV,thSV)  d=2: dicT(10, rrT,thT)
// d=3: dicTP(11, rrT,thT)   d=4: dsV_bug(20, rrSH,thSH)   (dsH == dicH)
// ---------------------------------------------------------------------------
__global__ void dict_kernel(const float* rrT, const float* thT,
                            const float* rrSH, const float* thSH,
                            const float* rrSV, const float* thSV,
                            float* dicH, float* dicV, float* dicT,
                            float* dicTP, float* dsV) {
  int tid = threadIdx.x;
  if (tid >= 5 * 41) return;
  int d = tid / 41, col = tid % 41;
  const float *rr, *th; float* W; int rows;
  switch (d) {
    case 0: rr = rrSH; th = thSH; W = dicH;  rows = 16; break;
    case 1: rr = rrSV; th = thSV; W = dicV;  rows = 20; break;
    case 2: rr = rrT;  th = thT;  W = dicT;  rows = 10; break;
    case 3: rr = rrT;  th = thT;  W = dicTP; rows = 11; break;
    default: rr = rrSH; th = thSH; W = dsV;  rows = 20; break;
  }
  float tmp[20]; float g2 = 0.f;
  for (int i = 0; i < rows; ++i) {
    float e;
    if (col == 0) e = 1.f;
    else {
      int pole = (col - 1) % 10, grp = (col - 1) / 10;
      float p = powf(rr[pole], (float)i);
      float ang = (float)i * th[pole];
      float sgn = (i & 1) ? -1.f : 1.f;
      if (grp == 0)      e = p * cosf(ang);
      else if (grp == 1) e = sgn * p * cosf(ang);
      else if (grp == 2) e = p * sinf(ang);
      else               e = sgn * p * sinf(ang);
    }
    tmp[i] = e; g2 += e * e;
  }
  float G = sqrtf(g2);
  if (G == 0.f) G = sqrtf((float)rows);
  for (int i = 0; i < rows; ++i) W[i * 41 + col] = tmp[i] / G;
}

// ---------------------------------------------------------------------------
// FISTA prep per dictionary: DtD, L=frobenius(DtD), linv, lam, A=I-DtD*linv
// padded to 48x48 with zeros.  params: [lamH,linvH, lamV,linvV, lamT,linvT]
// ---------------------------------------------------------------------------
__global__ void prep_kernel(const float* dicH, const float* dicV, const float* dicT,
                            float* A_H, float* A_V, float* A_T, float* params) {
  __shared__ float DtD[41 * 41];
  __shared__ float red[256];
  __shared__ float s_linv;
  int d = blockIdx.x;
  const float* D = (d == 0) ? dicH : (d == 1) ? dicV : dicT;
  int rows = (d == 0) ? 16 : (d == 1) ? 20 : 10;
  float* A = (d == 0) ? A_H : (d == 1) ? A_V : A_T;
  for (int p = threadIdx.x; p < 41 * 41; p += 256) {
    int i = p / 41, j = p % 41;
    float s = 0.f;
    for (int t = 0; t < rows; ++t) s += D[t * 41 + i] * D[t * 41 + j];
    DtD[p] = s;
  }
  __syncthreads();
  float ss = 0.f;
  for (int p = threadIdx.x; p < 41 * 41; p += 256) { float v = DtD[p]; ss += v * v; }
  red[threadIdx.x] = ss; __syncthreads();
  for (int o = 128; o > 0; o >>= 1) {
    if (threadIdx.x < o) red[threadIdx.x] += red[threadIdx.x + o];
    __syncthreads();
  }
  if (threadIdx.x == 0) {
    float L = sqrtf(red[0]);
    float linv = 1.f / L;
    s_linv = linv;
    params[d * 2 + 0] = LAMBD * linv;
    params[d * 2 + 1] = linv;
  }
  __syncthreads();
  float linv = s_linv;
  for (int p = threadIdx.x; p < KPAD * KPAD; p += 256) {
    int i = p / KPAD, j = p % KPAD;
    float v = 0.f;
    if (i < 41 && j < 41) v = ((i == j) ? 1.f : 0.f) - DtD[i * 41 + j] * linv;
    A[p] = v;
  }
}

__global__ void zero_kernel(float* p, long n) {
  long i = (long)blockIdx.x * blockDim.x + threadIdx.x;
  long stride = (long)gridDim.x * blockDim.x;
  for (; i < n; i += stride) p[i] = 0.f;
}

// ---------------------------------------------------------------------------
// DtY packs.  Spatial buffers: [48][NC] k-major, column layouts:
//  H: c = (((t*2+var)*16)+b)*160 + yi    V: c = (((t*2+var)*16)+b)*128 + xi
// ---------------------------------------------------------------------------
__global__ void pack_spatialH(const float* __restrict__ xin,
                              const float* __restrict__ dicH,
                              const float* __restrict__ params,
                              float* __restrict__ DtY) {
  int tid = blockIdx.x * blockDim.x + threadIdx.x;
  if (tid >= KPAD * NC_H) return;
  int c = tid % NC_H;
  int k = tid / NC_H;
  float out = 0.f;
  if (k < 41) {
    int yi = c % 160, g = c / 160;
    int b = g & 15, tv = g >> 4, t = tv >> 1, var = tv & 1;
    int bb = b >> 3, chunk = b & 7;
    float linv = params[1];
    float s = 0.f;
    for (int w = 0; w < 16; ++w) {
      int xi = chunk * 16 + w;
      if (var) xi = 127 - xi;
      s += dicH[w * 41 + k] * xin[bb * 204800 + t * 20480 + xi * 160 + yi];
    }
    out = s * linv;
  }
  DtY[tid] = out;
}

__global__ void pack_spatialV(const float* __restrict__ xin,
                              const float* __restrict__ dicV,
                              const float* __restrict__ params,
                              float* __restrict__ DtY) {
  int tid = blockIdx.x * blockDim.x + threadIdx.x;
  if (tid >= KPAD * NC_V) return;
  int c = tid % NC_V;
  int k = tid / NC_V;
  float out = 0.f;
  if (k < 41) {
    int xi = c % 128, g = c / 128;
    int b = g & 15, tv = g >> 4, t = tv >> 1, var = tv & 1;
    int bb = b >> 3, c8 = b & 7;
    float linv = params[3];
    float s = 0.f;
    for (int w = 0; w < 20; ++w) {
      int yi = c8 * 20 + w;
      if (var) yi = 159 - yi;
      s += dicV[w * 41 + k] * xin[bb * 204800 + t * 20480 + xi * 160 + yi];
    }
    out = s * linv;
  }
  DtY[tid] = out;
}

// Temporal DtY from spatial FISTA results (never materializes 'out'):
// DtY_T[k][b*FEAT+m] = linvT * sum_t dicT[t][k] * spatial_x(t, b, m)
__global__ void pack_temporal(const float* __restrict__ Xh,
                              const float* __restrict__ Xv,
                              const float* __restrict__ dicT,
                              const float* __restrict__ params,
                              float* __restrict__ DtY) {
  int tid = blockIdx.x * blockDim.x + threadIdx.x;
  if (tid >= KPAD * NC_T) return;   // 18.1M < 2^31
  int m = tid % FEAT;
  int r = tid / FEAT;
  int b = r & 15;
  int k = r >> 4;
  float out = 0.f;
  if (k < 41) {
    int type, var, ks, ml;
    if (m < OFF_V0)      { type = 0; var = 0; ks = m / 160;              ml = m % 160; }
    else if (m < OFF_H1) { type = 1; var = 0; ks = (m - OFF_V0) / 128;   ml = (m - OFF_V0) % 128; }
    else if (m < OFF_V1) { type = 0; var = 1; ks = (m - OFF_H1) / 160;   ml = (m - OFF_H1) % 160; }
    else                 { type = 1; var = 1; ks = (m - OFF_V1) / 128;   ml = (m - OFF_V1) % 128; }
    float linv = params[5];
    float s = 0.f;
    for (int t = 0; t < 10; ++t) {
      float v;
      if (type == 0) v = Xh[ks * NC_H + ((((t * 2 + var) * 16) + b) * 160 + ml)];
      else           v = Xv[ks * NC_V + ((((t * 2 + var) * 16) + b) * 128 + ml)];
      s += dicT[t * 41 + k] * v;
    }
    out = s * linv;
  }
  DtY[tid] = out;
}

// ---------------------------------------------------------------------------
// FISTA iteration:  x = softshrink(A@y + DtY, lam);  y = x + tt*(x - x_old)
// One wave per 16-column tile; 48x16 output tile via 3x12 V_WMMA_F32_16X16X4_F32.
// NC is a template parameter so all row offsets fold to 32-bit constants
// (no 64-bit address multiplies in the hot loop).
// Grids are exact multiples of waves -> EXEC always all-ones (WMMA requirement).
// ---------------------------------------------------------------------------
template <int NC>
__global__ void fista_iter(const float* __restrict__ A48,
                           const float* __restrict__ params, int lamIdx,
                           const float* __restrict__ DtY,
                           float* __restrict__ x, float* __restrict__ y,
                           float tt) {
  const int lane = threadIdx.x & 31;
  const int wave = ((int)blockIdx.x * (int)blockDim.x + (int)threadIdx.x) >> 5;
  const int lrow = lane & 15;
  const int half = lane >> 4;          // 0 or 1
  const int col = wave * 16 + lrow;    // this lane's column

  // Preload A tiles: a[kt][j4] -> rows kt*16+lrow, cols j4*4 + 2*half + {0,1}
  v2f a[3][12];
#pragma unroll
  for (int kt = 0; kt < 3; ++kt) {
#pragma unroll
    for (int j4 = 0; j4 < 12; ++j4) {
      const float* ap = A48 + (kt * 16 + lrow) * KPAD + j4 * 4 + 2 * half;
      v2f t; t[0] = ap[0]; t[1] = ap[1];
      a[kt][j4] = t;
    }
  }

  v8f acc[3];
#pragma unroll
  for (int kt = 0; kt < 3; ++kt)
#pragma unroll
    for (int v = 0; v < 8; ++v) acc[kt][v] = 0.f;

  // B operand: rows jr = j4*4 + 2*half + {0,1}; constant-offset loads from ybase.
  const float* ybase = y + (2 * half) * NC + col;
#pragma unroll
  for (int j4 = 0; j4 < 12; ++j4) {
    v2f bv;
    bv[0] = ybase[(j4 * 4) * NC];
    bv[1] = ybase[(j4 * 4 + 1) * NC];
    acc[0] = __builtin_amdgcn_wmma_f32_16x16x4_f32(false, a[0][j4], false, bv,
                                                   (short)0, acc[0], false, false);
    acc[1] = __builtin_amdgcn_wmma_f32_16x16x4_f32(false, a[1][j4], false, bv,
                                                   (short)0, acc[1], false, false);
    acc[2] = __builtin_amdgcn_wmma_f32_16x16x4_f32(false, a[2][j4], false, bv,
                                                   (short)0, acc[2], false, false);
  }

  const float lam = params[lamIdx];
  // C/D layout: element (k = kt*16 + v + 8*half, col)
  const int ebase = (8 * half) * NC + col;
  const float* dty = DtY + ebase;
  float* xp = x + ebase;
  float* yp = y + ebase;
#pragma unroll
  for (int kt = 0; kt < 3; ++kt) {
#pragma unroll
    for (int v = 0; v < 8; ++v) {
      const int off = (kt * 16 + v) * NC;     // compile-time constant * NC
      float z = acc[kt][v] + dty[off];
      float az = fabsf(z) - lam;
      float xn = (az > 0.f) ? copysignf(az, z) : 0.f;
      float xo = xp[off];
      xp[off] = xn;
      yp[off] = xn + tt * (xn - xo);
    }
  }
}

// cPred[b][m] = sum_k dicTP[10][k] * c[b][k][m]
__global__ void cpred_kernel(const float* __restrict__ Xt,
                             const float* __restrict__ dicTP,
                             float* __restrict__ cPred) {
  int tid = blockIdx.x * blockDim.x + threadIdx.x;
  if (tid >= 16 * FEAT) return;
  float s = 0.f;
  for (int k = 0; k < 41; ++k)
    s += dicTP[10 * 41 + k] * Xt[k * NC_T + tid];
  cPred[tid] = s;
}

// Final reconstruction: 4 outputs, each (2,128,160), concatenated in d_out.
__global__ void recon_kernel(const float* __restrict__ cPred,
                             const float* __restrict__ dsH,   // == dicH
                             const float* __restrict__ dsV,   // bug: rrSH params
                             float* __restrict__ out) {
  int tid = blockIdx.x * blockDim.x + threadIdx.x;
  if (tid >= 4 * 40960) return;
  int sect = tid / 40960, rem = tid % 40960;
  int bb = rem / 20480, r2 = rem % 20480;
  int xi = r2 / 160, yi = r2 % 160;
  float s = 0.f;
  if (sect == 0) {
    int bp = bb * 8 + xi / 16, w = xi % 16;
    const float* cp = cPred + bp * FEAT + OFF_H0 + yi;
    for (int k = 0; k < 41; ++k) s += dsH[w * 41 + k] * cp[k * 160];
  } else if (sect == 1) {
    int bp = bb * 8 + yi / 20, w = yi % 20;
    const float* cp = cPred + bp * FEAT + OFF_V0 + xi;
    for (int k = 0; k < 41; ++k) s += dsV[w * 41 + k] * cp[k * 128];
  } else if (sect == 2) {
    int X = 127 - xi;
    int bp = bb * 8 + X / 16, w = X % 16;
    const float* cp = cPred + bp * FEAT + OFF_H1 + yi;
    for (int k = 0; k < 41; ++k) s += dsH[w * 41 + k] * cp[k * 160];
  } else {
    int Yr = 159 - yi;
    int bp = bb * 8 + Yr / 20, w = Yr % 20;
    const float* cp = cPred + bp * FEAT + OFF_V1 + xi;
    for (int k = 0; k < 41; ++k) s += dsV[w * 41 + k] * cp[k * 128];
  }
  out[tid] = s;
}

// ---------------------------------------------------------------------------
extern "C" void kernel_launch(void* const* d_in, const int* in_sizes, int n_in,
                              void* d_out, int out_size, void* d_ws, size_t ws_size,
                              hipStream_t stream) {
  const float* xin = (const float*)d_in[0];
  const float* rrT  = (const float*)d_in[1];
  const float* thT  = (const float*)d_in[2];
  const float* rrSH = (const float*)d_in[3];
  const float* thSH = (const float*)d_in[4];
  const float* rrSV = (const float*)d_in[5];
  const float* thSV = (const float*)d_in[6];

  float* ws = (float*)d_ws;
  size_t off = 0;
  auto alloc = [&](size_t n) { float* p = ws + off; off += (n + 63) & ~(size_t)63; return p; };

  float* dicH  = alloc(16 * 41);
  float* dicV  = alloc(20 * 41);
  float* dicT  = alloc(10 * 41);
  float* dicTP = alloc(11 * 41);
  float* dsV   = alloc(20 * 41);
  float* A_H = alloc(KPAD * KPAD);
  float* A_V = alloc(KPAD * KPAD);
  float* A_T = alloc(KPAD * KPAD);
  float* params = alloc(16);
  const long LEN_H = (long)KPAD * NC_H;
  const long LEN_V = (long)KPAD * NC_V;
  const long LEN_T = (long)KPAD * NC_T;
  float* DtY_H = alloc(LEN_H); float* x_H = alloc(LEN_H); float* y_H = alloc(LEN_H);
  float* DtY_V = alloc(LEN_V); float* x_V = alloc(LEN_V); float* y_V = alloc(LEN_V);
  float* DtY_T = alloc(LEN_T); float* x_T = alloc(LEN_T); float* y_T = alloc(LEN_T);
  float* cPred = alloc((long)16 * FEAT);

  dict_kernel<<<1, 256, 0, stream>>>(rrT, thT, rrSH, thSH, rrSV, thSV,
                                     dicH, dicV, dicT, dicTP, dsV);
  prep_kernel<<<3, 256, 0, stream>>>(dicH, dicV, dicT, A_H, A_V, A_T, params);

  // zero x/y state (x and y are contiguous per stage; lengths are 64-float multiples)
  zero_kernel<<<2048, 256, 0, stream>>>(x_H, 2 * LEN_H);
  zero_kernel<<<2048, 256, 0, stream>>>(x_V, 2 * LEN_V);
  zero_kernel<<<2048, 256, 0, stream>>>(x_T, 2 * LEN_T);

  pack_spatialH<<<(int)(LEN_H / 256), 256, 0, stream>>>(xin, dicH, params, DtY_H);
  pack_spatialV<<<(int)(LEN_V / 256), 256, 0, stream>>>(xin, dicV, params, DtY_V);

  // spatial FISTAs (H and V share the momentum schedule; both start at t=1)
  double t = 1.0;
  for (int it = 0; it < 100; ++it) {
    double tn = (1.0 + sqrt(1.0 + 4.0 * t * t)) * 0.5;
    float tt = (float)((t - 1.0) / tn);
    t = tn;
    fista_iter<NC_H><<<(NC_H / 16) / 8, 256, 0, stream>>>(A_H, params, 0, DtY_H, x_H, y_H, tt);
    fista_iter<NC_V><<<(NC_V / 16) / 8, 256, 0, stream>>>(A_V, params, 2, DtY_V, x_V, y_V, tt);
  }

  pack_temporal<<<(int)(LEN_T / 256), 256, 0, stream>>>(x_H, x_V, dicT, params, DtY_T);

  t = 1.0;
  for (int it = 0; it < 100; ++it) {
    double tn = (1.0 + sqrt(1.0 + 4.0 * t * t)) * 0.5;
    float tt = (float)((t - 1.0) / tn);
    t = tn;
    fista_iter<NC_T><<<(NC_T / 16) / 8, 256, 0, stream>>>(A_T, params, 4, DtY_T, x_T, y_T, tt);
  }

  cpred_kernel<<<(16 * FEAT) / 256, 256, 0, stream>>>(x_T, dicTP, cPred);
  recon_kernel<<<(4 * 40960) / 256, 256, 0, stream>>>(cPred, dicH, dsV, (float*)d_out);
}